// Model_21114059227276
// MI455X (gfx1250) — compile-verified
//
#include <hip/hip_runtime.h>

// Problem dimensions (fixed by the reference).
#define B_DIM 128
#define S_DIM 128
#define G_DIM 8
#define D_DIM 1024

typedef float v4f __attribute__((ext_vector_type(4)));
typedef int   v4i __attribute__((ext_vector_type(4)));

#define AS1 __attribute__((address_space(1)))
#define AS3 __attribute__((address_space(3)))

#if defined(__has_builtin)
#if __has_builtin(__builtin_amdgcn_global_load_async_to_lds_b128)
#define USE_ASYNC_LDS 1
#endif
#endif

// One workgroup per (b,s) position. 256 threads * float4 = one D=1024 row.
// Gather <=8 embedding rows (L2-resident 8MB table), average, add pos row.
__global__ __launch_bounds__(256) void embed_phon_kernel(
    const int*   __restrict__ phon_tokens,    // [B,S,G] int32
    const int*   __restrict__ group_len_raw,  // [B,S]   int32 (true len = raw+1)
    const float* __restrict__ phon_emb,       // [VOCAB, D] f32
    const float* __restrict__ pos_emb,        // [S, D]     f32
    float*       __restrict__ out)            // [B,S,D]    f32
{
    const int bs  = blockIdx.x;          // 0 .. B*S-1
    const int s   = bs & (S_DIM - 1);
    const int tid = threadIdx.x;         // 0 .. 255
    const int col = tid << 2;            // 4 consecutive floats per thread

    __shared__ v4f lds_pos[D_DIM / 4];   // 4 KB staging for the pos-emb row

    // ---- Stage the position-embedding row into LDS asynchronously so the
    // ---- copy overlaps the gather below (CDNA5 async-to-LDS path).
    const float* pos_src = pos_emb + (size_t)s * D_DIM + col;
#ifdef USE_ASYNC_LDS
    __builtin_amdgcn_global_load_async_to_lds_b128(
        (AS1 v4i*)pos_src, (AS3 v4i*)&lds_pos[tid], 0, 0);
#else
    lds_pos[tid] = *(const v4f*)pos_src;
#endif

    // ---- Wave-uniform group metadata (blockIdx-only indexing -> SMEM loads).
    const int  len  = group_len_raw[bs] + 1;        // in [1, G]
    const int* toks = phon_tokens + bs * G_DIM;

    // Branchless slot indices: invalid slots (g >= len) re-read the LAST
    // VALID slot's row -> identical per-lane addresses -> WGP$ hits, zero
    // extra L2 traffic. All selects are wave-uniform scalar ops.
    int tok[G_DIM];
#pragma unroll
    for (int g = 0; g < G_DIM; ++g) {
        const int gi = (g < len) ? g : (len - 1);
        tok[g] = toks[gi];
    }

    // ---- Issue all 8 b128 gathers back-to-back (single clause, ONE
    // ---- load-latency exposure instead of ~4.5 serialized round-trips).
    v4f val[G_DIM];
#pragma unroll
    for (int g = 0; g < G_DIM; ++g) {
        val[g] = *(const v4f*)(phon_emb + (size_t)tok[g] * D_DIM + col);
    }

    // ---- Weighted accumulate: w in {0,1}. fma(v,1,acc) == acc+v exactly;
    // ---- fma(v,0,acc) == acc for finite table values. Two accumulators
    // ---- halve the FMA dependency chain.
    v4f acc0 = (v4f)(0.0f);
    v4f acc1 = (v4f)(0.0f);
#pragma unroll
    for (int g = 0; g < G_DIM; g += 2) {
        const float w0 = (g     < len) ? 1.0f : 0.0f;
        const float w1 = (g + 1 < len) ? 1.0f : 0.0f;
        acc0 += val[g]     * w0;
        acc1 += val[g + 1] * w1;
    }
    v4f acc = acc0 + acc1;

    const float inv = 1.0f / (float)len;

    // ---- Fence the async copy (same-lane LDS bytes; no barrier needed).
#ifdef USE_ASYNC_LDS
    asm volatile("s_wait_asynccnt 0" ::: "memory");
#endif
    const v4f p = lds_pos[tid];          // ds_load_b128

    v4f res = acc * inv + p;

    // ---- Non-temporal b128 store: the 64MB output streams to HBM without
    // ---- evicting the L2-resident embedding table.
    __builtin_nontemporal_store(res, (v4f*)(out + (size_t)bs * D_DIM + col));
}

extern "C" void kernel_launch(void* const* d_in, const int* in_sizes, int n_in,
                              void* d_out, int out_size, void* d_ws, size_t ws_size,
                              hipStream_t stream) {
    const int*   phon_tokens   = (const int*)d_in[0];
    const int*   group_len_raw = (const int*)d_in[1];
    const float* phon_emb      = (const float*)d_in[2];
    const float* pos_emb       = (const float*)d_in[3];
    float*       out           = (float*)d_out;

    dim3 grid(B_DIM * S_DIM);   // 16384 workgroups
    dim3 block(D_DIM / 4);      // 256 threads = 8 wave32s

    embed_phon_kernel<<<grid, block, 0, stream>>>(
        phon_tokens, group_len_raw, phon_emb, pos_emb, out);
}